// Net2_79087527788732
// MI455X (gfx1250) — compile-verified
//
#include <hip/hip_runtime.h>
#include <math.h>

#define N_NODES 200000
#define N_EDGES 5000000
#define D_IN    128
#define HID     16
#define D_OUT   2

typedef __attribute__((ext_vector_type(2))) float v2f;
typedef __attribute__((ext_vector_type(8))) float v8f;

// ---------------------------------------------------------------------------
// Degree / normalization:  deg[i] = 1 (self loop) + #incoming edges
// dis[i] = rsqrt(deg[i])   (deg >= 1 always, so the where() is trivial)
// ---------------------------------------------------------------------------
__global__ void k_init_deg(float* __restrict__ deg) {
    int i = blockIdx.x * blockDim.x + threadIdx.x;
    if (i < N_NODES) deg[i] = 1.0f;
}

__global__ void k_accum_deg(const int* __restrict__ dst, float* __restrict__ deg) {
    int e = blockIdx.x * blockDim.x + threadIdx.x;
    if (e < N_EDGES) atomicAdd(&deg[dst[e]], 1.0f);
}

__global__ void k_rsqrt(float* __restrict__ dis) {
    int i = blockIdx.x * blockDim.x + threadIdx.x;
    if (i < N_NODES) dis[i] = rsqrtf(dis[i]);
}

// ---------------------------------------------------------------------------
// WMMA f32 GEMM:  out[N_NODES x NOUT] = A[N_NODES x K] @ W[K x NOUT]
// One wave32 computes a 16-row x 16-col tile via V_WMMA_F32_16X16X4_F32,
// looping K in steps of 4.  NOUT < 16 pads B with zeros and masks stores.
// ISA f32 layouts: A lane L: row M=L&15, holds K = k+2*(L>=16) .. +1
//                  B lane L: col N=L&15, holds same K pair
//                  C/D VGPR r: M = r + 8*(L>=16), N = L&15
// ---------------------------------------------------------------------------
template<int K, int NOUT>
__global__ void k_gemm_wmma(const float* __restrict__ A,
                            const float* __restrict__ W,
                            float* __restrict__ out) {
    const int NTILES = N_NODES / 16;                       // 12500 exactly
    int wave = (blockIdx.x * blockDim.x + threadIdx.x) >> 5;
    int tile = wave < NTILES ? wave : (NTILES - 1);        // clamp: EXEC stays all-1s
    int lane = threadIdx.x & 31;
    int half = lane >> 4;            // 0: K pair (k,k+1) / rows M0..7 ; 1: (k+2,k+3) / M8..15
    int mn   = lane & 15;            // row for A, col for B/D
    const int row_base = tile * 16;

    v8f c = (v8f)0.0f;
    const float* arow = A + (long)(row_base + mn) * K;
#pragma unroll
    for (int k = 0; k < K; k += 4) {
        int kk = k + half * 2;
        v2f a, b;
        a.x = arow[kk];
        a.y = arow[kk + 1];
        if (NOUT == 16) {
            b.x = W[kk * 16 + mn];
            b.y = W[(kk + 1) * 16 + mn];
        } else {
            b.x = (mn < NOUT) ? W[kk * NOUT + mn] : 0.0f;
            b.y = (mn < NOUT) ? W[(kk + 1) * NOUT + mn] : 0.0f;
        }
        c = __builtin_amdgcn_wmma_f32_16x16x4_f32(
                /*neg_a=*/false, a, /*neg_b=*/false, b,
                /*c_mod=*/(short)0, c, /*reuse_a=*/false, /*reuse_b=*/false);
    }

    if (mn < NOUT) {
#pragma unroll
        for (int r = 0; r < 8; ++r) {
            int m = r + half * 8;
            out[(long)(row_base + m) * NOUT + mn] = c[r];
        }
    }
}

// ---------------------------------------------------------------------------
// Aggregation:  agg[i] = dis[i]^2 * hpre[i]  (self loop, atomic-free init)
//               then edge scatter: agg[dst] += dis[src]*dis[dst] * hpre[src]
// ---------------------------------------------------------------------------
template<int F>
__global__ void k_selfloop_init(const float* __restrict__ hpre,
                                const float* __restrict__ dis,
                                float* __restrict__ agg) {
    int t = blockIdx.x * blockDim.x + threadIdx.x;
    if (t < N_NODES * F) {
        int i = (F == 16) ? (t >> 4) : (t >> 1);
        float d = dis[i];
        agg[t] = hpre[t] * d * d;
    }
}

// 4 threads per edge, float4 gather, 4 scalar f32 global atomics each.
__global__ void k_edge_scatter16(const int* __restrict__ src,
                                 const int* __restrict__ dst,
                                 const float* __restrict__ dis,
                                 const float* __restrict__ hpre,
                                 float* __restrict__ agg) {
    long t = (long)blockIdx.x * blockDim.x + threadIdx.x;
    if (t >= (long)N_EDGES * 4) return;
    int e = (int)(t >> 2), q = (int)(t & 3);
    int s = src[e], d = dst[e];
    float nrm = dis[s] * dis[d];
    float4 v = *(const float4*)(hpre + (long)s * 16 + q * 4);
    float* ap = agg + (long)d * 16 + q * 4;
    atomicAdd(ap + 0, v.x * nrm);
    atomicAdd(ap + 1, v.y * nrm);
    atomicAdd(ap + 2, v.z * nrm);
    atomicAdd(ap + 3, v.w * nrm);
}

// 1 thread per edge, 2 features.
__global__ void k_edge_scatter2(const int* __restrict__ src,
                                const int* __restrict__ dst,
                                const float* __restrict__ dis,
                                const float* __restrict__ hpre,
                                float* __restrict__ agg) {
    int e = blockIdx.x * blockDim.x + threadIdx.x;
    if (e >= N_EDGES) return;
    int s = src[e], d = dst[e];
    float nrm = dis[s] * dis[d];
    float2 v = *(const float2*)(hpre + (long)s * 2);
    atomicAdd(agg + (long)d * 2 + 0, v.x * nrm);
    atomicAdd(agg + (long)d * 2 + 1, v.y * nrm);
}

__global__ void k_bias_relu16(const float* __restrict__ agg,
                              const float* __restrict__ b,
                              float* __restrict__ out) {
    int t = blockIdx.x * blockDim.x + threadIdx.x;
    if (t < N_NODES * 16) {
        float v = agg[t] + b[t & 15];
        out[t] = v > 0.0f ? v : 0.0f;
    }
}

__global__ void k_logsoftmax2(const float* __restrict__ agg,
                              const float* __restrict__ b2,
                              float* __restrict__ out) {
    int i = blockIdx.x * blockDim.x + threadIdx.x;
    if (i >= N_NODES) return;
    float z0 = agg[i * 2 + 0] + b2[0];
    float z1 = agg[i * 2 + 1] + b2[1];
    float m  = fmaxf(z0, z1);
    float lse = m + logf(expf(z0 - m) + expf(z1 - m));
    out[i * 2 + 0] = z0 - lse;
    out[i * 2 + 1] = z1 - lse;
}

// ---------------------------------------------------------------------------
extern "C" void kernel_launch(void* const* d_in, const int* in_sizes, int n_in,
                              void* d_out, int out_size, void* d_ws, size_t ws_size,
                              hipStream_t stream) {
    const float* x   = (const float*)d_in[0];
    const int*   ei  = (const int*)  d_in[1];   // [2, E]: row0 = src, row1 = dst
    const float* W1  = (const float*)d_in[2];
    const float* b1  = (const float*)d_in[3];
    const float* W3  = (const float*)d_in[4];
    const float* b3  = (const float*)d_in[5];
    const float* W2  = (const float*)d_in[6];
    const float* b2  = (const float*)d_in[7];
    float* out = (float*)d_out;

    const int* src = ei;
    const int* dst = ei + N_EDGES;

    // Workspace layout (floats), total ~26.4 MB
    float* ws   = (float*)d_ws;
    float* dis  = ws;                         // N
    float* bufA = ws + 200704;                // N*16 (16B aligned)
    float* bufB = bufA + (long)N_NODES * HID; // N*16

    const int B = 256;
    const int gN    = (N_NODES + B - 1) / B;            // 782
    const int gE    = (N_EDGES + B - 1) / B;            // 19532
    const int gW    = (N_NODES / 16 + 7) / 8;           // 1563 blocks, 8 waves each
    const int gN16  = (N_NODES * 16 + B - 1) / B;       // 12500
    const int gN2   = (N_NODES * 2 + B - 1) / B;        // 1563
    const int gE4   = (int)(((long)N_EDGES * 4 + B - 1) / B); // 78125

    // --- normalization ---
    k_init_deg <<<gN, B, 0, stream>>>(dis);
    k_accum_deg<<<gE, B, 0, stream>>>(dst, dis);
    k_rsqrt    <<<gN, B, 0, stream>>>(dis);

    // --- conv1: x@W1 -> aggregate -> relu(+b1) ---
    k_gemm_wmma<D_IN, HID><<<gW, B, 0, stream>>>(x, W1, bufA);
    k_selfloop_init<16>   <<<gN16, B, 0, stream>>>(bufA, dis, bufB);
    k_edge_scatter16      <<<gE4, B, 0, stream>>>(src, dst, dis, bufA, bufB);
    k_bias_relu16         <<<gN16, B, 0, stream>>>(bufB, b1, bufA);   // h1 in bufA

    // --- conv3: h1@W3 -> aggregate -> relu(+b3) ---
    k_gemm_wmma<HID, HID><<<gW, B, 0, stream>>>(bufA, W3, bufB);      // hpre in bufB
    k_selfloop_init<16>  <<<gN16, B, 0, stream>>>(bufB, dis, bufA);   // agg in bufA
    k_edge_scatter16     <<<gE4, B, 0, stream>>>(src, dst, dis, bufB, bufA);
    k_bias_relu16        <<<gN16, B, 0, stream>>>(bufA, b3, bufB);    // h3 in bufB

    // --- conv2: h3@W2 -> aggregate -> log_softmax(+b2) ---
    k_gemm_wmma<HID, D_OUT><<<gW, B, 0, stream>>>(bufB, W2, bufA);    // hpre2 in bufA (N*2)
    k_selfloop_init<2>     <<<gN2, B, 0, stream>>>(bufA, dis, bufB);  // agg2 in bufB (N*2)
    k_edge_scatter2        <<<gE, B, 0, stream>>>(src, dst, dis, bufA, bufB);
    k_logsoftmax2          <<<gN, B, 0, stream>>>(bufB, b2, out);
}